// GraphConvolution_87213605913035
// MI455X (gfx1250) — compile-verified
//
#include <hip/hip_runtime.h>
#include <hip/hip_bf16.h>

typedef __attribute__((ext_vector_type(2))) float v2f;
typedef __attribute__((ext_vector_type(8))) float v8f;

#define N_NODES 50000
#define E_TOT   800000
#define ACT_C   1.6791f
#define INV_SQRT3 0.57735026918962576f
#define INV_SQRT32 0.17677669529663687f
#define INV_SQRT96 0.10206207261596575f

// ---------------------------------------------------------------------------
// Kernel 0: zero the segment-sum scratch (N*384 seg + N cnt floats)
// ---------------------------------------------------------------------------
__global__ void zero_ws_kernel(float* __restrict__ p, int n) {
    int i = blockIdx.x * blockDim.x + threadIdx.x;
    if (i < n) p[i] = 0.0f;
}

// ---------------------------------------------------------------------------
// Kernel 1: fused edge MLP (WMMA f32) + tensor product + scatter-add
// 64 threads = 2 waves per block; each wave owns a 16-edge tile.
// ---------------------------------------------------------------------------
__global__ __launch_bounds__(64) void edge_kernel(
    const float* __restrict__ x,            // N x 160
    const float* __restrict__ edge_scalars, // E x 64
    const float* __restrict__ edge_attr_sh, // E x 4
    const int*   __restrict__ edge_index,   // 2 x E
    const float* __restrict__ w_fc1,        // 64 x 64
    const float* __restrict__ w_fc2,        // 64 x 192
    float* __restrict__ seg,                // N x 384
    float* __restrict__ cnt)                // N
{
    __shared__ float hs[2][16 * 66];     // h tile, padded stride (bank-conflict free b64 reads)
    __shared__ float wsm[2][16 * 192];   // weight tile

    const int wave = threadIdx.x >> 5;
    const int lane = threadIdx.x & 31;
    const int half = lane >> 4;          // 0: lanes 0-15, 1: lanes 16-31
    const int lrow = lane & 15;
    const int e0   = (blockIdx.x * 2 + wave) * 16;

    // ---------------- GEMM1: h16x64 = edge_scalars[tile] (16x64) @ w_fc1 (64x64)
    v8f acc1[4] = {};
    for (int k = 0; k < 16; ++k) {
        const int colb = 4 * k + 2 * half;                 // A frag: K pair base
        v2f a = *(const v2f*)(edge_scalars + (size_t)(e0 + lrow) * 64 + colb);
#pragma unroll
        for (int t = 0; t < 4; ++t) {
            const int c = 16 * t + lrow;
            v2f b;
            b.x = w_fc1[(size_t)colb * 64 + c];
            b.y = w_fc1[(size_t)(colb + 1) * 64 + c];
            acc1[t] = __builtin_amdgcn_wmma_f32_16x16x4_f32(
                false, a, false, b, (short)0, acc1[t], false, false);
        }
    }

    // silu(z/8)*ACT_C via fast v_rcp_f32 (avoid IEEE div_scale/fixup chains),
    // store to LDS row-major (transpose D-layout -> A-layout)
    float* myhs = hs[wave];
#pragma unroll
    for (int t = 0; t < 4; ++t) {
#pragma unroll
        for (int j = 0; j < 8; ++j) {
            const int m = j + 8 * half;
            const int c = 16 * t + lrow;
            float z = acc1[t][j] * 0.125f;               // 1/sqrt(64)
            float sig = __builtin_amdgcn_rcpf(1.0f + __expf(-z));
            myhs[m * 66 + c] = z * sig * ACT_C;          // silu * act const
        }
    }
    __syncthreads();

    // ---------------- GEMM2: weight16x192 = h (16x64) @ w_fc2 (64x192)
    v8f acc2[12] = {};
    for (int k = 0; k < 16; ++k) {
        const int r = 4 * k + 2 * half;
        v2f a = *(const v2f*)(myhs + lrow * 66 + r);
#pragma unroll
        for (int t = 0; t < 12; ++t) {
            const int c = 16 * t + lrow;
            v2f b;
            b.x = w_fc2[(size_t)r * 192 + c];
            b.y = w_fc2[(size_t)(r + 1) * 192 + c];
            acc2[t] = __builtin_amdgcn_wmma_f32_16x16x4_f32(
                false, a, false, b, (short)0, acc2[t], false, false);
        }
    }

    float* myw = wsm[wave];
#pragma unroll
    for (int t = 0; t < 12; ++t) {
#pragma unroll
        for (int j = 0; j < 8; ++j) {
            myw[(j + 8 * half) * 192 + 16 * t + lrow] = acc2[t][j] * 0.125f; // 1/sqrt(64)
        }
    }
    __syncthreads();

    // ---------------- tensor product + scatter (12 branch-uniform passes/edge)
    for (int e = 0; e < 16; ++e) {
        const int ge  = e0 + e;
        const int src = edge_index[ge];
        const int dst = edge_index[E_TOT + ge];
        const float sh0 = edge_attr_sh[(size_t)ge * 4 + 0];
        const float sh1 = edge_attr_sh[(size_t)ge * 4 + 1];
        const float sh2 = edge_attr_sh[(size_t)ge * 4 + 2];
        const float sh3 = edge_attr_sh[(size_t)ge * 4 + 3];
        const float* xr = x + (size_t)src * 160;
        const float* wr = myw + e * 192;
        float* sr = seg + (size_t)dst * 384;
#pragma unroll
        for (int p = 0; p < 12; ++p) {
            const int c = p * 32 + lane;
            float val;
            if (c < 64) {                 // mid_s0: w0 * x_s * sh_s
                val = wr[c] * xr[c] * sh0;
            } else if (c < 96) {          // mid_s1: w3 * dot(x_v, sh_v)/sqrt3
                const int u = c - 64;
                const float* gv = xr + 64 + u * 3;
                val = wr[160 + u] * (gv[0] * sh1 + gv[1] * sh2 + gv[2] * sh3) * INV_SQRT3;
            } else if (c < 288) {         // mid_v0: w1 * x_s * sh_v
                const int q = c - 96;
                const int u = q / 3, i = q - 3 * u;
                const float shv = (i == 0) ? sh1 : ((i == 1) ? sh2 : sh3);
                val = wr[64 + u] * xr[u] * shv;
            } else {                      // mid_v1: w2 * x_v * sh_s
                const int q = c - 288;
                const int u = q / 3, i = q - 3 * u;
                val = wr[128 + u] * xr[64 + u * 3 + i] * sh0;
            }
            atomicAdd(sr + c, val);
        }
    }
    if (lane < 16) {
        const int dst = edge_index[E_TOT + e0 + lane];
        atomicAdd(cnt + dst, 1.0f);
    }
}

// ---------------------------------------------------------------------------
// Kernel 2: per-node mean + alpha-gated linear + self-connection
// one block (64 threads) per node
// ---------------------------------------------------------------------------
__global__ __launch_bounds__(64) void node_kernel(
    const float* __restrict__ x,        // N x 160
    const float* __restrict__ seg,      // N x 384
    const float* __restrict__ cnt,      // N
    const float* __restrict__ w_sc_s,   // 64 x 64
    const float* __restrict__ w_sc_v,   // 32 x 32
    const float* __restrict__ w_lin_s,  // 96 x 64
    const float* __restrict__ w_lin_v,  // 96 x 32
    const float* __restrict__ w_alpha,  // 96 x 1
    float* __restrict__ out)            // N x 160
{
    const int n = blockIdx.x;
    const int t = threadIdx.x;
    __shared__ float mean[384];
    __shared__ float xrow[160];

    const float inv = 1.0f / fmaxf(cnt[n], 1.0f);
    const float* sr = seg + (size_t)n * 384;
    for (int i = t; i < 384; i += 64) mean[i] = sr[i] * inv;
    for (int i = t; i < 160; i += 64) xrow[i] = x[(size_t)n * 160 + i];
    __syncthreads();

    // alpha (redundantly per thread; broadcast LDS reads)
    float al = 0.0f;
    for (int u = 0; u < 96; ++u) al += mean[u] * w_alpha[u];
    al *= INV_SQRT96;

    // scalar channel: out[n, t], t < 64
    {
        float ls = 0.0f, sc = 0.0f;
        for (int u = 0; u < 96; ++u) ls += mean[u] * w_lin_s[u * 64 + t];
        for (int u = 0; u < 64; ++u) sc += xrow[u] * w_sc_s[u * 64 + t];
        out[(size_t)n * 160 + t] = sc * 0.125f + al * ls * INV_SQRT96;
    }

    // vector channel: 96 outputs over 2 passes
#pragma unroll
    for (int pass = 0; pass < 2; ++pass) {
        const int idx = pass * 64 + t;
        if (idx < 96) {
            const int w = idx / 3, i = idx - 3 * w;
            float lv = 0.0f, sv = 0.0f;
            for (int u = 0; u < 96; ++u) lv += mean[96 + u * 3 + i] * w_lin_v[u * 32 + w];
            for (int u = 0; u < 32; ++u) sv += xrow[64 + u * 3 + i] * w_sc_v[u * 32 + w];
            out[(size_t)n * 160 + 64 + idx] = sv * INV_SQRT32 + al * lv * INV_SQRT96;
        }
    }
}

// ---------------------------------------------------------------------------
extern "C" void kernel_launch(void* const* d_in, const int* in_sizes, int n_in,
                              void* d_out, int out_size, void* d_ws, size_t ws_size,
                              hipStream_t stream) {
    (void)in_sizes; (void)n_in; (void)out_size; (void)ws_size;
    const float* x            = (const float*)d_in[0];
    const float* edge_scalars = (const float*)d_in[1];
    const float* edge_attr_sh = (const float*)d_in[2];
    const int*   edge_index   = (const int*)  d_in[3];
    const float* w_fc1        = (const float*)d_in[4];
    const float* w_fc2        = (const float*)d_in[5];
    const float* w_sc_s       = (const float*)d_in[6];
    const float* w_sc_v       = (const float*)d_in[7];
    const float* w_lin_s      = (const float*)d_in[8];
    const float* w_lin_v      = (const float*)d_in[9];
    const float* w_alpha      = (const float*)d_in[10];
    float* out = (float*)d_out;

    float* seg = (float*)d_ws;                 // N*384
    float* cnt = seg + (size_t)N_NODES * 384;  // N

    const int zn = N_NODES * 385;
    zero_ws_kernel<<<(zn + 255) / 256, 256, 0, stream>>>(seg, zn);

    // 2 waves/block, 16 edges/wave -> 32 edges/block; 800000/32 = 25000 blocks exactly
    edge_kernel<<<E_TOT / 32, 64, 0, stream>>>(
        x, edge_scalars, edge_attr_sh, edge_index, w_fc1, w_fc2, seg, cnt);

    node_kernel<<<N_NODES, 64, 0, stream>>>(
        x, seg, cnt, w_sc_s, w_sc_v, w_lin_s, w_lin_v, w_alpha, out);
}